// GCN_67903432950131
// MI455X (gfx1250) — compile-verified
//
#include <hip/hip_runtime.h>
#include <hip/hip_bf16.h>

#define N_NODES 100000
#define N_EDGES 1200000
#define IN_DIM  256
#define HID     64

typedef float v2f __attribute__((ext_vector_type(2)));
typedef float v8f __attribute__((ext_vector_type(8)));

__device__ __forceinline__ float selu_f(float x) {
    const float alpha = 1.6732632423543772f;
    const float scale = 1.0507009873554805f;
    return scale * (x > 0.0f ? x : alpha * (__expf(x) - 1.0f));
}

// ---------------------------------------------------------------------------
// Dense GEMM  Out[M x 64] = f(A[M x K]) * W[K x 64]   using V_WMMA_F32_16X16X4_F32
// f = identity, or selu(a + bias[k]) when FUSE_SELU (bias over the K dim).
// One wave per 16-row M tile; 4 N-tiles of 16 -> 4 v8f accumulators (A reuse x4).
//
// W is staged into LDS pre-swizzled in k-pair-interleaved order:
//   sW[(k>>1)*128 + col*2 + (k&1)] = W[k][col]
// so each B fragment {W[ka][col], W[ka+1][col]} is a single contiguous 8-byte
// read -> one ds_load_b64 into an even-aligned VGPR pair (no v_mov shuffles).
// M must be a multiple of 16 (100000 = 6250*16).
// ---------------------------------------------------------------------------
template <int K, bool FUSE_SELU>
__launch_bounds__(256)
__global__ void gemm_wmma_n64(const float* __restrict__ A,
                              const float* __restrict__ W,
                              const float* __restrict__ bias,
                              float* __restrict__ Out, int M)
{
    __shared__ float sW[K * 64];
    for (int i = threadIdx.x; i < K * 64; i += 256) {
        const int k   = i >> 6;
        const int col = i & 63;
        sW[(k >> 1) * 128 + col * 2 + (k & 1)] = W[i];
    }
    __syncthreads();

    const int lane  = threadIdx.x & 31;
    const int wave  = threadIdx.x >> 5;
    const int tile  = blockIdx.x * 8 + wave;
    const int ntile = M >> 4;
    if (tile >= ntile) return;          // wave-uniform: EXEC stays all-1s for WMMA

    const int  r0  = lane & 15;         // row (A) / col (B,C) within tile
    const int  kq  = lane >> 4;         // 0 or 1 (selects K pair 0-1 vs 2-3)
    const long row = (long)tile * 16 + r0;
    const float* __restrict__ arow = A + row * (long)K;

    v8f acc0 = {}, acc1 = {}, acc2 = {}, acc3 = {};

    for (int kk = 0; kk < K; kk += 4) {
        const int ka = kk + kq * 2;
        // A fragment: VGPR0 = A[row][ka], VGPR1 = A[row][ka+1]  (ISA 16x4 f32 layout)
        v2f a;
        if constexpr (FUSE_SELU) {
            a.x = selu_f(arow[ka]     + bias[ka]);
            a.y = selu_f(arow[ka + 1] + bias[ka + 1]);
        } else {
            a = *(const v2f*)(arow + ka);           // 8B-aligned: ka even, row 256B-aligned
        }
        // B fragments: one b64 LDS read each, 128B apart across the 4 N-tiles
        const float* wp = &sW[(ka >> 1) * 128 + r0 * 2];
        const v2f b0 = *(const v2f*)(wp + 0);
        const v2f b1 = *(const v2f*)(wp + 32);
        const v2f b2 = *(const v2f*)(wp + 64);
        const v2f b3 = *(const v2f*)(wp + 96);
        acc0 = __builtin_amdgcn_wmma_f32_16x16x4_f32(false, a, false, b0, (short)0, acc0, false, false);
        acc1 = __builtin_amdgcn_wmma_f32_16x16x4_f32(false, a, false, b1, (short)0, acc1, false, false);
        acc2 = __builtin_amdgcn_wmma_f32_16x16x4_f32(false, a, false, b2, (short)0, acc2, false, false);
        acc3 = __builtin_amdgcn_wmma_f32_16x16x4_f32(false, a, false, b3, (short)0, acc3, false, false);
    }

    // C/D layout: VGPR r -> M = r (lanes 0-15) or r+8 (lanes 16-31), N = r0
    const int rowhi = (lane >> 4) * 8;
    float* __restrict__ outp = Out + ((long)tile * 16 + rowhi) * 64 + r0;
#pragma unroll
    for (int r = 0; r < 8; ++r) {
        outp[r * 64 + 0]  = acc0[r];
        outp[r * 64 + 16] = acc1[r];
        outp[r * 64 + 32] = acc2[r];
        outp[r * 64 + 48] = acc3[r];
    }
}

// ---------------------------------------------------------------------------
// Zero workspace accumulator (float4 stores)
// ---------------------------------------------------------------------------
__global__ void zero_f32v4(float4* __restrict__ p, int n4)
{
    int i = blockIdx.x * blockDim.x + threadIdx.x;
    if (i < n4) p[i] = make_float4(0.f, 0.f, 0.f, 0.f);
}

// ---------------------------------------------------------------------------
// SpMM scatter: Z[rows[e]] += vals[e] * H[cols[e]].  16 threads/edge, float4 each.
// Gathers hit L2 (H = 25.6 MB << 192 MB); scatter via global_atomic_add_f32.
// ---------------------------------------------------------------------------
__launch_bounds__(256)
__global__ void spmm_scatter(const int* __restrict__ rows,
                             const int* __restrict__ cols,
                             const float* __restrict__ vals,
                             const float* __restrict__ H,
                             float* __restrict__ Z, int nEdges)
{
    const int tid = blockIdx.x * blockDim.x + threadIdx.x;
    const int e   = tid >> 4;
    const int s   = tid & 15;
    if (e >= nEdges) return;

    const int   r = rows[e];
    const int   c = cols[e];
    const float v = vals[e];

    const float4 h = ((const float4*)(H + (size_t)c * 64))[s];
    float* zp = Z + (size_t)r * 64 + s * 4;
    atomicAdd(zp + 0, v * h.x);
    atomicAdd(zp + 1, v * h.y);
    atomicAdd(zp + 2, v * h.z);
    atomicAdd(zp + 3, v * h.w);
}

// ---------------------------------------------------------------------------
// out[node] = (Z[node] + b2) / max(||Z[node]+b2||_2, 1e-12).  One wave32/node.
// ---------------------------------------------------------------------------
__launch_bounds__(256)
__global__ void bias_l2norm(const float* __restrict__ Z,
                            const float* __restrict__ b,
                            float* __restrict__ out, int nNodes)
{
    const int tid  = blockIdx.x * blockDim.x + threadIdx.x;
    const int node = tid >> 5;
    const int lane = tid & 31;
    if (node >= nNodes) return;

    float2 z = ((const float2*)(Z + (size_t)node * 64))[lane];
    z.x += b[lane * 2];
    z.y += b[lane * 2 + 1];

    float s = z.x * z.x + z.y * z.y;
#pragma unroll
    for (int off = 16; off > 0; off >>= 1) s += __shfl_xor(s, off, 32);

    const float inv = 1.0f / fmaxf(sqrtf(s), 1e-12f);
    ((float2*)(out + (size_t)node * 64))[lane] = make_float2(z.x * inv, z.y * inv);
}

// ---------------------------------------------------------------------------
// Launch: inputs = X1, X2, rows1, cols1, vals1, rows2, cols2, vals2, W1, b1, W2, b2
// Output = Z1 (100000x64) || Z2 (100000x64), float32.
// Workspace: two ping-pong 25.6 MB buffers.
// ---------------------------------------------------------------------------
extern "C" void kernel_launch(void* const* d_in, const int* in_sizes, int n_in,
                              void* d_out, int out_size, void* d_ws, size_t ws_size,
                              hipStream_t stream)
{
    const float* X[2]    = { (const float*)d_in[0], (const float*)d_in[1] };
    const int*   rows[2] = { (const int*)d_in[2],   (const int*)d_in[5] };
    const int*   cols[2] = { (const int*)d_in[3],   (const int*)d_in[6] };
    const float* vals[2] = { (const float*)d_in[4], (const float*)d_in[7] };
    const float* W1 = (const float*)d_in[8];
    const float* b1 = (const float*)d_in[9];
    const float* W2 = (const float*)d_in[10];
    const float* b2 = (const float*)d_in[11];
    float* out = (float*)d_out;

    float* buf0 = (float*)d_ws;                       // H / H2
    float* buf1 = buf0 + (size_t)N_NODES * HID;       // Z accumulators

    const int gemm_blocks = (N_NODES / 16 + 7) / 8;           // 782
    const int zero_blocks = (N_NODES * HID / 4 + 255) / 256;  // 6250
    const int spmm_blocks = (N_EDGES * 16) / 256;             // 75000
    const int norm_blocks = (N_NODES * 32) / 256;             // 12500

    for (int g = 0; g < 2; ++g) {
        // 1) H = X * W1
        gemm_wmma_n64<IN_DIM, false><<<gemm_blocks, 256, 0, stream>>>(
            X[g], W1, nullptr, buf0, N_NODES);
        // 2) Z1acc = A @ H
        zero_f32v4<<<zero_blocks, 256, 0, stream>>>((float4*)buf1, N_NODES * HID / 4);
        spmm_scatter<<<spmm_blocks, 256, 0, stream>>>(
            rows[g], cols[g], vals[g], buf0, buf1, N_EDGES);
        // 3) H2 = selu(Z1acc + b1) * W2   (bias+SELU fused into A-load)
        gemm_wmma_n64<HID, true><<<gemm_blocks, 256, 0, stream>>>(
            buf1, W2, b1, buf0, N_NODES);
        // 4) Z2acc = A @ H2
        zero_f32v4<<<zero_blocks, 256, 0, stream>>>((float4*)buf1, N_NODES * HID / 4);
        spmm_scatter<<<spmm_blocks, 256, 0, stream>>>(
            rows[g], cols[g], vals[g], buf0, buf1, N_EDGES);
        // 5) out = normalize(Z2acc + b2)
        bias_l2norm<<<norm_blocks, 256, 0, stream>>>(
            buf1, b2, out + (size_t)g * N_NODES * HID, N_NODES);
    }
}